// GraphRNN_24111946400619
// MI455X (gfx1250) — compile-verified
//
#include <hip/hip_runtime.h>
#include <math.h>

typedef __attribute__((ext_vector_type(2))) float v2f;
typedef __attribute__((ext_vector_type(8))) float v8f;

#define HDIM 64
#define PDIM 5
#define G3H  192
#define WPAD 200   // LDS row stride (words): 2*WPAD mod 64 = 16 -> disjoint bank groups per half-wave

// fp32 WMMA: D(16x16) = A(16x4) * B(4x16) + C  -> v_wmma_f32_16x16x4_f32
__device__ __forceinline__ v8f wmma4(v2f a, v2f b, v8f c) {
  return __builtin_amdgcn_wmma_f32_16x16x4_f32(
      /*neg_a=*/false, a, /*neg_b=*/false, b,
      /*c_mod=*/(short)0, c, /*reuse_a=*/false, /*reuse_b=*/false);
}

__device__ __forceinline__ float sigmoidf_(float x) {
  return 1.0f / (1.0f + expf(-x));
}

// ---------------- degree / normalization ----------------
__global__ void k_deg_init(float* __restrict__ deg, int n) {
  int i = blockIdx.x * blockDim.x + threadIdx.x;
  if (i < n) deg[i] = 1.0f;  // self-loop weight
}

__global__ void k_deg_acc(const float* __restrict__ ew, const int* __restrict__ dst,
                          float* __restrict__ deg, int e) {
  int i = blockIdx.x * blockDim.x + threadIdx.x;
  if (i < e) atomicAdd(&deg[dst[i]], ew[i]);
}

__global__ void k_dinv(float* __restrict__ deg, int n) {
  int i = blockIdx.x * blockDim.x + threadIdx.x;
  if (i < n) deg[i] = rsqrtf(deg[i]);
}

// per-edge norm, shared by both GCN layers
__global__ void k_norm(const float* __restrict__ ew, const int* __restrict__ src,
                       const int* __restrict__ dst, const float* __restrict__ dinv,
                       float* __restrict__ norm, int e) {
  int i = blockIdx.x * blockDim.x + threadIdx.x;
  if (i < e) norm[i] = dinv[src[i]] * ew[i] * dinv[dst[i]];
}

// ---------------- layer-1 scatter: rank-1 xw computed on the fly ----------------
__global__ void k_scatter1(const float* __restrict__ x, const float* __restrict__ W1,
                           const int* __restrict__ src, const int* __restrict__ dst,
                           const float* __restrict__ norm, float* agg, int nedges) {
  long long tid = (long long)blockIdx.x * blockDim.x + threadIdx.x;
  int e = (int)(tid >> 5);
  if (e >= nedges) return;
  int t = (int)(tid & 31);
  int s = src[e], d = dst[e];
  float c = x[s] * norm[e];                      // broadcast within wave
  const float2 w = *(const float2*)(W1 + 2 * t); // L1-resident
  float* ap = agg + (size_t)d * HDIM + 2 * t;
  atomicAdd(ap + 0, w.x * c);
  atomicAdd(ap + 1, w.y * c);
}

// ---------------- layer-2 scatter: gather dense xw rows ----------------
__global__ void k_scatter2(const float* __restrict__ xw, const int* __restrict__ src,
                           const int* __restrict__ dst, const float* __restrict__ norm,
                           float* agg, int nedges) {
  long long tid = (long long)blockIdx.x * blockDim.x + threadIdx.x;
  int e = (int)(tid >> 5);
  if (e >= nedges) return;
  int t = (int)(tid & 31);
  int s = src[e], d = dst[e];
  float nm = norm[e];
  const float2 v = *(const float2*)(xw + (size_t)s * HDIM + 2 * t);
  float* ap = agg + (size_t)d * HDIM + 2 * t;
  atomicAdd(ap + 0, v.x * nm);
  atomicAdd(ap + 1, v.y * nm);
}

// ---------------- fused layer-1 finalize + layer-2 GEMM ----------------
// A(row,k) = relu(agg1[row,k] + dinv[row]^2 * x[row]*W1[k] + b1[k]);  Y = A @ W2
__global__ void k_gemm64_f(const float* __restrict__ agg, const float* __restrict__ x,
                           const float* __restrict__ W1, const float* __restrict__ b1,
                           const float* __restrict__ dinv, const float* __restrict__ W2,
                           float* __restrict__ Y, int nrows) {
  int lane = threadIdx.x & 31;
  int wave = blockIdx.x * (blockDim.x >> 5) + (threadIdx.x >> 5);
  int row0 = wave << 4;
  if (row0 >= nrows) return;  // wave-uniform (EXEC stays all-ones for WMMA)
  int half = lane >> 4;
  int m = lane & 15;

  float di = dinv[row0 + m];
  float cx = di * di * x[row0 + m];
  const float* arow = agg + (size_t)(row0 + m) * HDIM;

  v8f a0 = {}, a1 = {}, a2 = {}, a3 = {};
  for (int k0 = 0; k0 < HDIM; k0 += 4) {
    int kk = k0 + 2 * half;
    float2 ag = *(const float2*)(arow + kk);
    v2f a;
    a.x = fmaxf(ag.x + cx * W1[kk] + b1[kk], 0.0f);
    a.y = fmaxf(ag.y + cx * W1[kk + 1] + b1[kk + 1], 0.0f);
    const float* w0 = W2 + (size_t)kk * HDIM + m;
    const float* w1 = w0 + HDIM;
    v2f b0; b0.x = w0[0];  b0.y = w1[0];
    v2f b1v; b1v.x = w0[16]; b1v.y = w1[16];
    v2f b2v; b2v.x = w0[32]; b2v.y = w1[32];
    v2f b3v; b3v.x = w0[48]; b3v.y = w1[48];
    a0 = wmma4(a, b0, a0);
    a1 = wmma4(a, b1v, a1);
    a2 = wmma4(a, b2v, a2);
    a3 = wmma4(a, b3v, a3);
  }
  int rbase = half ? 8 : 0;
#pragma unroll
  for (int r = 0; r < 8; ++r) {
    float* yrow = Y + (size_t)(row0 + rbase + r) * HDIM + m;
    yrow[0]  = a0[r];
    yrow[16] = a1[r];
    yrow[32] = a2[r];
    yrow[48] = a3[r];
  }
}

// ---------------- transpose [192x64] -> [64x192] for coalesced staging ----------------
__global__ void k_transpose(const float* __restrict__ w, float* __restrict__ wT, int total) {
  int i = blockIdx.x * blockDim.x + threadIdx.x;
  if (i >= total) return;
  int j = i / HDIM, k = i % HDIM;   // w[j][k], j<192, k<64
  wT[k * G3H + j] = w[i];
}

// ---------------- fused layer-2 finalize + GRU cell (weights staged in LDS) ----------------
__global__ void k_gru_f(const float* __restrict__ agg, const float* __restrict__ xw,
                        const float* __restrict__ b2, const float* __restrict__ dinv,
                        const float* __restrict__ hin,
                        const float* __restrict__ wihT, const float* __restrict__ whhT,
                        const float* __restrict__ bih, const float* __restrict__ bhh,
                        float* __restrict__ hout, int nrows) {
  extern __shared__ float smem[];
  float* swi = smem;                 // [64][WPAD]
  float* swh = smem + HDIM * WPAD;   // [64][WPAD]
  for (int i = threadIdx.x; i < HDIM * G3H; i += blockDim.x) {
    int k = i / G3H, j = i % G3H;
    swi[k * WPAD + j] = wihT[i];
    swh[k * WPAD + j] = whhT[i];
  }
  __syncthreads();

  int lane = threadIdx.x & 31;
  int wave = blockIdx.x * (blockDim.x >> 5) + (threadIdx.x >> 5);
  int row0 = wave << 4;
  if (row0 >= nrows) return;
  int half = lane >> 4;
  int m = lane & 15;

  float di = dinv[row0 + m];
  float di2 = di * di;
  const float* arow = agg + (size_t)(row0 + m) * HDIM;
  const float* xrow = xw + (size_t)(row0 + m) * HDIM;
  const float* hrow = hin + (size_t)(row0 + m) * HDIM;

  v8f zero = {};
  v8f gi[12], gh[12];
#pragma unroll
  for (int c = 0; c < 12; ++c) { gi[c] = zero; gh[c] = zero; }

  for (int k0 = 0; k0 < HDIM; k0 += 4) {
    int kk = k0 + 2 * half;
    float2 ag = *(const float2*)(arow + kk);
    float2 xv = *(const float2*)(xrow + kk);
    v2f ax;
    ax.x = fmaxf(ag.x + di2 * xv.x + b2[kk], 0.0f);      // xf = relu(GCN2 finalize)
    ax.y = fmaxf(ag.y + di2 * xv.y + b2[kk + 1], 0.0f);
    float2 hv2 = *(const float2*)(hrow + kk);
    v2f ah; ah.x = hv2.x; ah.y = hv2.y;
    const float* wi0 = swi + kk * WPAD + m;
    const float* wi1 = wi0 + WPAD;
    const float* wh0 = swh + kk * WPAD + m;
    const float* wh1 = wh0 + WPAD;
#pragma unroll
    for (int c = 0; c < 12; ++c) {
      v2f bi; bi.x = wi0[c * 16]; bi.y = wi1[c * 16];
      gi[c] = wmma4(ax, bi, gi[c]);
      v2f bh; bh.x = wh0[c * 16]; bh.y = wh1[c * 16];
      gh[c] = wmma4(ah, bh, gh[c]);
    }
  }

  int rbase = half ? 8 : 0;
#pragma unroll
  for (int c = 0; c < 4; ++c) {
    int col = c * 16 + m;
    float br_i = bih[col],       br_h = bhh[col];
    float bz_i = bih[64 + col],  bz_h = bhh[64 + col];
    float bn_i = bih[128 + col], bn_h = bhh[128 + col];
#pragma unroll
    for (int r = 0; r < 8; ++r) {
      int row = row0 + rbase + r;
      float rg = sigmoidf_(gi[c][r] + br_i + gh[c][r] + br_h);
      float zg = sigmoidf_(gi[c + 4][r] + bz_i + gh[c + 4][r] + bz_h);
      float ng = tanhf(gi[c + 8][r] + bn_i + rg * (gh[c + 8][r] + bn_h));
      float hv = hin[(size_t)row * HDIM + col];
      hout[(size_t)row * HDIM + col] = (1.0f - zg) * ng + zg * hv;
    }
  }
}

// ---------------- fc head: out[N x 5] = h_new @ fc_w + fc_b ----------------
__global__ void k_fc(const float* __restrict__ hnew, const float* __restrict__ fcw,
                     const float* __restrict__ fcb, float* __restrict__ out, int n) {
  int i = blockIdx.x * blockDim.x + threadIdx.x;
  if (i >= n) return;
  const float* hr = hnew + (size_t)i * HDIM;
  float acc[PDIM];
#pragma unroll
  for (int p = 0; p < PDIM; ++p) acc[p] = fcb[p];
  for (int h = 0; h < HDIM; ++h) {
    float hv = hr[h];
#pragma unroll
    for (int p = 0; p < PDIM; ++p) acc[p] += hv * fcw[h * PDIM + p];
  }
#pragma unroll
  for (int p = 0; p < PDIM; ++p) out[(size_t)i * PDIM + p] = acc[p];
}

static inline int cdiv(long long a, long long b) { return (int)((a + b - 1) / b); }

extern "C" void kernel_launch(void* const* d_in, const int* in_sizes, int n_in,
                              void* d_out, int out_size, void* d_ws, size_t ws_size,
                              hipStream_t stream) {
  const float* x    = (const float*)d_in[0];
  const float* ew   = (const float*)d_in[1];
  const float* h    = (const float*)d_in[2];
  const float* W1   = (const float*)d_in[3];
  const float* b1   = (const float*)d_in[4];
  const float* W2   = (const float*)d_in[5];
  const float* b2   = (const float*)d_in[6];
  const float* w_ih = (const float*)d_in[7];
  const float* w_hh = (const float*)d_in[8];
  const float* b_ih = (const float*)d_in[9];
  const float* b_hh = (const float*)d_in[10];
  const float* fc_w = (const float*)d_in[11];
  const float* fc_b = (const float*)d_in[12];
  const int*   ei   = (const int*)d_in[13];

  const int E  = in_sizes[1];
  const int NH = in_sizes[2];           // N * 64
  const int N  = NH / HDIM;
  const int* src = ei;                  // edge_index[0]
  const int* dst = ei + E;              // edge_index[1]

  float* ws   = (float*)d_ws;
  float* dinv = ws;                      // N
  float* norm = dinv + N;                // E
  float* bufA = norm + (size_t)E;        // N*64  (xw2)
  float* bufB = bufA + (size_t)NH;       // N*64  (agg1 / agg2)
  float* wihT = bufB + (size_t)NH;       // 64*192
  float* whhT = wihT + HDIM * G3H;       // 64*192

  float* out  = (float*)d_out;                 // N*5
  float* hout = out + (size_t)N * PDIM;        // N*64

  const int TB = 256;

  // degrees -> dinv -> per-edge norm
  k_deg_init<<<cdiv(N, TB), TB, 0, stream>>>(dinv, N);
  k_deg_acc <<<cdiv(E, TB), TB, 0, stream>>>(ew, dst, dinv, E);
  k_dinv    <<<cdiv(N, TB), TB, 0, stream>>>(dinv, N);
  k_norm    <<<cdiv(E, TB), TB, 0, stream>>>(ew, src, dst, dinv, norm, E);

  // transpose GRU weights once (coalesced LDS staging in k_gru_f)
  k_transpose<<<cdiv(G3H * HDIM, TB), TB, 0, stream>>>(w_ih, wihT, G3H * HDIM);
  k_transpose<<<cdiv(G3H * HDIM, TB), TB, 0, stream>>>(w_hh, whhT, G3H * HDIM);

  // ---- GCN layer 1: scatter with on-the-fly rank-1 xw ----
  hipMemsetAsync(bufB, 0, (size_t)NH * sizeof(float), stream);
  k_scatter1<<<cdiv((long long)E * 32, TB), TB, 0, stream>>>(x, W1, src, dst, norm, bufB, E);

  // ---- fused relu(finalize1) + GEMM: bufA = xf1 @ W2 ----
  k_gemm64_f<<<cdiv(N, 128), TB, 0, stream>>>(bufB, x, W1, b1, dinv, W2, bufA, N);

  // ---- GCN layer 2: scatter dense rows ----
  hipMemsetAsync(bufB, 0, (size_t)NH * sizeof(float), stream);
  k_scatter2<<<cdiv((long long)E * 32, TB), TB, 0, stream>>>(bufA, src, dst, norm, bufB, E);

  // ---- fused relu(finalize2) + GRU cell -> h_new ----
  size_t lds_bytes = (size_t)2 * HDIM * WPAD * sizeof(float);  // 102400 B
  k_gru_f<<<cdiv(N / 16, 8), TB, lds_bytes, stream>>>(bufB, bufA, b2, dinv, h,
                                                      wihT, whhT, b_ih, b_hh, hout, N);

  // ---- fc head ----
  k_fc<<<cdiv(N, TB), TB, 0, stream>>>(hout, fc_w, fc_b, out, N);
}